// position_encode_69080253989611
// MI455X (gfx1250) — compile-verified
//
#include <hip/hip_runtime.h>
#include <hip/hip_bf16.h>
#include <math.h>

typedef float v2f __attribute__((ext_vector_type(2)));
typedef float v8f __attribute__((ext_vector_type(8)));

#define BATCH   8
#define NPTS    4096
#define CFEAT   64
#define KOUT    16      // K neighbors kept
#define DIL     2       // dilation
#define TOPK    (KOUT * DIL)   // 32 kept during selection

#define BQ      128     // queries per block
#define WQ      16      // queries per wave (one WMMA M-tile)
#define NWAVES  8       // 256 threads
#define CT      16      // candidates per tile (one WMMA N-tile)
#define NTILES  (NPTS / CT)    // 256 (even -> unroll-by-2 is safe)

// ---------------------------------------------------------------------------
// Kernel 0: per-point squared norm of the 64-dim feature rows.
// ---------------------------------------------------------------------------
__global__ __launch_bounds__(256) void sqnorm_kernel(const float* __restrict__ pts,
                                                     float* __restrict__ sqn) {
    int i = blockIdx.x * blockDim.x + threadIdx.x;      // 0 .. B*N-1
    const float* r = pts + (size_t)i * CFEAT;
    float s = 0.0f;
#pragma unroll
    for (int c = 0; c < CFEAT; c += 4) {
        float4 v = *(const float4*)(r + c);
        s += v.x * v.x + v.y * v.y + v.z * v.z + v.w * v.w;
    }
    sqn[i] = s;
}

// ---------------------------------------------------------------------------
// Kernel 1: fused X·X^T (WMMA f32 16x16x4) + running top-32 selection + dilate.
// Each wave handles 16 queries; block = 8 waves = 128 queries.
// B fragments double-buffered in registers: loads for tile ct+1 fly while the
// wmma chain consumes tile ct.
// ---------------------------------------------------------------------------
__global__ __launch_bounds__(256) void knn_kernel(const float* __restrict__ pts,
                                                  const float* __restrict__ sqn,
                                                  int* __restrict__ nnidx) {
    __shared__ float sDist[NWAVES][WQ * CT];            //  8 KB
    __shared__ float sTopD[NWAVES][WQ][TOPK];           // 16 KB
    __shared__ int   sTopI[NWAVES][WQ][TOPK];           // 16 KB

    const int tid  = threadIdx.x;
    const int wid  = tid >> 5;
    const int lane = tid & 31;
    const int l    = lane & 15;     // row/col within 16-tile
    const int hi   = lane >> 4;     // which K-half (A/B) / M-half (C)

    const int b  = blockIdx.y;
    const int q0 = blockIdx.x * BQ + wid * WQ;

    const float* P  = pts + (size_t)b * NPTS * CFEAT;
    const float* SQ = sqn + (size_t)b * NPTS;

    // Preload A fragments (16x4 f32 layout): lane l holds query row q0+l,
    // hi selects K={0,1} vs K={2,3} of each 4-wide chunk. Fold the -2 here.
    v2f afrag[16];
    {
        const float* arow = P + (size_t)(q0 + l) * CFEAT + 2 * hi;
#pragma unroll
        for (int c = 0; c < 16; ++c) {
            v2f t = *(const v2f*)(arow + 4 * c);
            afrag[c].x = -2.0f * t.x;
            afrag[c].y = -2.0f * t.y;
        }
    }

    // Init per-query top-32 lists (sorted ascending; [31] = current worst).
    if (hi == 0) {
#pragma unroll
        for (int i = 0; i < TOPK; ++i) {
            sTopD[wid][l][i] = INFINITY;
            sTopI[wid][l][i] = 0;
        }
    }
    asm volatile("s_wait_dscnt 0" ::: "memory");

    // ---- load one candidate tile's B fragments into a register buffer ----
    auto loadB = [&](v2f (&bf)[16], int cbase) {
        const float* brow = P + (size_t)(cbase + l) * CFEAT + 2 * hi;
#pragma unroll
        for (int c = 0; c < 16; ++c) bf[c] = *(const v2f*)(brow + 4 * c);
    };

    // ---- wmma + distance spill + top-32 insertion for one tile ----
    auto process = [&](const v2f (&bf)[16], int cbase, float sqc) {
        v8f acc = {};
#pragma unroll
        for (int c = 0; c < 16; ++c) {
            acc = __builtin_amdgcn_wmma_f32_16x16x4_f32(
                false, afrag[c], false, bf[c], (short)0, acc, false, false);
        }
        // d = ||c||^2 - 2 q.c   (query norm dropped: constant per row)
#pragma unroll
        for (int j = 0; j < 8; ++j) {
            const int q = hi * 8 + j;                   // C/D layout: vgpr j, half hi
            sDist[wid][q * CT + l] = acc[j] + sqc;
        }
        asm volatile("s_wait_dscnt 0" ::: "memory");    // wave-internal LDS ordering

        // Lanes 0..15: each owns one query row; insertion-maintain sorted top-32.
        if (hi == 0) {
            float* td = &sTopD[wid][l][0];
            int*   ti = &sTopI[wid][l][0];
            const float* dr = &sDist[wid][l * CT];
#pragma unroll 4
            for (int c = 0; c < CT; ++c) {
                const float d = dr[c];
                if (d < td[TOPK - 1]) {
                    int p = TOPK - 1;
                    while (p > 0 && td[p - 1] > d) {
                        td[p] = td[p - 1];
                        ti[p] = ti[p - 1];
                        --p;
                    }
                    td[p] = d;
                    ti[p] = cbase + c;
                }
            }
        }
        asm volatile("s_wait_dscnt 0" ::: "memory");
    };

    // Software pipeline, unrolled by 2 so both buffers use constant indices
    // (keeps them register-resident).
    v2f b0[16], b1[16];
    float sq0 = SQ[l];
    float sq1;
    loadB(b0, 0);

    for (int ct = 0; ct < NTILES; ct += 2) {
        // prefetch tile ct+1 while tile ct computes
        loadB(b1, (ct + 1) * CT);
        sq1 = SQ[(ct + 1) * CT + l];
        process(b0, ct * CT, sq0);

        // prefetch tile ct+2 while tile ct+1 computes
        if (ct + 2 < NTILES) {
            loadB(b0, (ct + 2) * CT);
            sq0 = SQ[(ct + 2) * CT + l];
        }
        process(b1, (ct + 1) * CT, sq1);
    }

    // Dilated take: sorted positions 0,2,...,30.
    if (hi == 0) {
        int* op = nnidx + ((size_t)b * NPTS + (q0 + l)) * KOUT;
#pragma unroll
        for (int k = 0; k < KOUT; ++k) op[k] = sTopI[wid][l][2 * k];
    }
}

// ---------------------------------------------------------------------------
// Kernel 2: gather neighbor xyz, build [dist | rel(3) | self(3) | nbr(3)].
// One thread per (b, n, k) output row (10 floats).
// ---------------------------------------------------------------------------
__global__ __launch_bounds__(256) void gather_kernel(const float* __restrict__ xyz,
                                                     const int* __restrict__ nnidx,
                                                     float* __restrict__ out) {
    int t = blockIdx.x * blockDim.x + threadIdx.x;      // 0 .. B*N*K-1
    const int bn = t / KOUT;
    const int n  = bn % NPTS;
    const int b  = bn / NPTS;

    const float* X = xyz + (size_t)b * NPTS * 3;
    const int idx = nnidx[t];

    const float qx = X[(size_t)n * 3 + 0];
    const float qy = X[(size_t)n * 3 + 1];
    const float qz = X[(size_t)n * 3 + 2];
    const float nx = X[(size_t)idx * 3 + 0];
    const float ny = X[(size_t)idx * 3 + 1];
    const float nz = X[(size_t)idx * 3 + 2];

    const float rx = qx - nx, ry = qy - ny, rz = qz - nz;
    const float d = sqrtf(rx * rx + ry * ry + rz * rz);

    float* o = out + (size_t)t * 10;
    o[0] = d;
    o[1] = rx; o[2] = ry; o[3] = rz;
    o[4] = qx; o[5] = qy; o[6] = qz;
    o[7] = nx; o[8] = ny; o[9] = nz;
}

// ---------------------------------------------------------------------------
extern "C" void kernel_launch(void* const* d_in, const int* in_sizes, int n_in,
                              void* d_out, int out_size, void* d_ws, size_t ws_size,
                              hipStream_t stream) {
    const float* new_xyz    = (const float*)d_in[0];   // [B, N, 3]
    const float* new_points = (const float*)d_in[1];   // [B, N, 64]
    float* out = (float*)d_out;                        // [B, N, K, 10]

    // Workspace layout: [sqnorm: B*N floats][nnidx: B*N*K ints]
    float* sqn   = (float*)d_ws;
    int*   nnidx = (int*)((char*)d_ws + (size_t)BATCH * NPTS * sizeof(float));

    // 0) squared norms of feature rows
    sqnorm_kernel<<<dim3((BATCH * NPTS) / 256), dim3(256), 0, stream>>>(new_points, sqn);

    // 1) fused WMMA distance GEMM + dilated kNN selection (double-buffered)
    knn_kernel<<<dim3(NPTS / BQ, BATCH), dim3(256), 0, stream>>>(new_points, sqn, nnidx);

    // 2) gather + position encoding
    gather_kernel<<<dim3((BATCH * NPTS * KOUT) / 256), dim3(256), 0, stream>>>(new_xyz, nnidx, out);
}